// NGCF_73280732004963
// MI455X (gfx1250) — compile-verified
//
#include <hip/hip_runtime.h>

#define DD 64
#define ET 16  // edges per tile in the edge-message kernel

typedef float v2f __attribute__((ext_vector_type(2)));
typedef float v8f __attribute__((ext_vector_type(8)));

// ---------------------------------------------------------------- utilities

__global__ void k_zero(float* __restrict__ p, int n) {
    int i = blockIdx.x * blockDim.x + threadIdx.x;
    if (i < n) p[i] = 0.0f;
}

__global__ void k_degree(const int* __restrict__ iu, const int* __restrict__ ii,
                         float* __restrict__ du, float* __restrict__ di, int E) {
    int e = blockIdx.x * blockDim.x + threadIdx.x;
    if (e < E) {
        atomicAdd(&du[iu[e]], 1.0f);
        atomicAdd(&di[ii[e]], 1.0f);
    }
}

__global__ void k_edge_w(const int* __restrict__ iu, const int* __restrict__ ii,
                         const float* __restrict__ du, const float* __restrict__ di,
                         float* __restrict__ w, int E) {
    int e = blockIdx.x * blockDim.x + threadIdx.x;
    if (e < E) w[e] = rsqrtf(du[iu[e]] * di[ii[e]]);
}

__global__ void k_gather_rows(const float* __restrict__ src, const int* __restrict__ idx,
                              float* __restrict__ dst, int nElem) {
    int i = blockIdx.x * blockDim.x + threadIdx.x;
    if (i < nElem) {
        int r = i >> 6, d = i & 63;
        dst[i] = src[(size_t)idx[r] * DD + d];
    }
}

__global__ void k_lrelu(float* __restrict__ x, int n) {
    int i = blockIdx.x * blockDim.x + threadIdx.x;
    if (i < n) {
        float v = x[i];
        x[i] = v > 0.0f ? v : 0.01f * v;
    }
}

// ------------------------------------------------------- WMMA node GEMM
// Y[N x 64] = X[N x 64] @ W[64 x 64]   (W row-major: W[k*64 + n])
// One wave computes one 16x16 output tile; K=64 -> 16 chained wmma f32 16x16x4.
__global__ __launch_bounds__(32) void k_gemm64(const float* __restrict__ X,
                                               const float* __restrict__ W,
                                               float* __restrict__ Y, int N) {
    const int lane = threadIdx.x;
    const int half = lane >> 4;     // 0: K pair {k,k+1}, 1: {k+2,k+3}
    const int l16  = lane & 15;
    const int m0 = blockIdx.x * 16;
    const int n0 = blockIdx.y * 16;

    int rowA = m0 + l16;
    if (rowA > N - 1) rowA = N - 1;               // clamp (keeps EXEC full)
    const float* xrow = X + (size_t)rowA * DD + half * 2;
    const float* wcol = W + (size_t)(half * 2) * DD + n0 + l16;

    v8f c = {0.f, 0.f, 0.f, 0.f, 0.f, 0.f, 0.f, 0.f};
#pragma unroll
    for (int kk = 0; kk < DD; kk += 4) {
        v2f a;
        a.x = xrow[kk];
        a.y = xrow[kk + 1];
        v2f b;
        b.x = wcol[kk * DD];
        b.y = wcol[kk * DD + DD];
        c = __builtin_amdgcn_wmma_f32_16x16x4_f32(false, a, false, b,
                                                  (short)0, c, false, false);
    }

    // C/D layout: VGPR r -> row m0 + r + half*8, col n0 + l16
#pragma unroll
    for (int r = 0; r < 8; ++r) {
        int row = m0 + half * 8 + r;
        if (row < N) Y[(size_t)row * DD + n0 + l16] = c[r];
    }
}

// ------------------------------------------------- WMMA edge message kernel
// For each edge e: z_int = (x_src[i_src[e]] * x_dst[i_dst[e]]) @ Wintr
//                  msg   = w[e] * (y_src[i_src[e]] + z_int)
//                  acc[i_dst[e]] += msg    (float atomics)
// 16 edges per block; 4 waves, each owning one 16-column tile.
__global__ __launch_bounds__(128) void k_edge_msg(
    const float* __restrict__ xsrc, const float* __restrict__ xdst,
    const int* __restrict__ isrc, const int* __restrict__ idst,
    const float* __restrict__ wE, const float* __restrict__ ysrc,
    const float* __restrict__ Wintr, float* __restrict__ acc, int E) {
    __shared__ float sP[ET][DD + 4];   // gathered elementwise products (A matrix)
    __shared__ float sZ[ET][DD + 4];   // gathered z_src rows
    __shared__ float sWv[ET];
    __shared__ int   sS[ET], sDt[ET];

    const int tid = threadIdx.x;
    const int e0 = blockIdx.x * ET;

    if (tid < ET) {
        int eg = e0 + tid;
        if (eg > E - 1) eg = E - 1;
        sS[tid]  = isrc[eg];
        sDt[tid] = idst[eg];
        sWv[tid] = wE[eg];
    }
    __syncthreads();

#pragma unroll
    for (int i = 0; i < (ET * DD) / 128; ++i) {
        int idx = tid + i * 128;
        int e = idx >> 6, d = idx & 63;
        size_t so = (size_t)sS[e] * DD + d;
        size_t dofs = (size_t)sDt[e] * DD + d;
        sP[e][d] = xsrc[so] * xdst[dofs];
        sZ[e][d] = ysrc[so];
    }
    __syncthreads();

    const int wave = tid >> 5;
    const int lane = tid & 31;
    const int half = lane >> 4;
    const int l16  = lane & 15;
    const int n0 = wave * 16;
    const float* wcol = Wintr + (size_t)(half * 2) * DD + n0 + l16;

    v8f c = {0.f, 0.f, 0.f, 0.f, 0.f, 0.f, 0.f, 0.f};
#pragma unroll
    for (int kk = 0; kk < DD; kk += 4) {
        v2f a;
        a.x = sP[l16][kk + half * 2];
        a.y = sP[l16][kk + half * 2 + 1];
        v2f b;
        b.x = wcol[kk * DD];
        b.y = wcol[kk * DD + DD];
        c = __builtin_amdgcn_wmma_f32_16x16x4_f32(false, a, false, b,
                                                  (short)0, c, false, false);
    }

#pragma unroll
    for (int r = 0; r < 8; ++r) {
        int eL = half * 8 + r;          // local edge (C-matrix row)
        int n  = n0 + l16;              // output feature
        if (e0 + eL < E) {
            float v = sWv[eL] * (sZ[eL][n] + c[r]);
            atomicAdd(&acc[(size_t)sDt[eL] * DD + n], v);
        }
    }
}

// ------------------------------------------------------------- final dot
// out[b] = sum_{s=0..2} dot(xu_s[pu[b]], xi_s[pi[b]])   (D=64 each stage)
// 16 lanes per pair, float4 loads, shfl reduction.
__global__ __launch_bounds__(256) void k_dot(
    const float* __restrict__ xu0, const float* __restrict__ xu1, const float* __restrict__ xu2,
    const float* __restrict__ xi0, const float* __restrict__ xi1, const float* __restrict__ xi2,
    const int* __restrict__ pu, const int* __restrict__ pi,
    float* __restrict__ out, int B) {
    int tid = threadIdx.x;
    int pair = blockIdx.x * (blockDim.x >> 4) + (tid >> 4);
    int l = tid & 15;
    if (pair >= B) return;
    size_t u = (size_t)pu[pair] * DD;
    size_t it = (size_t)pi[pair] * DD;
    const float* us[3] = {xu0 + u, xu1 + u, xu2 + u};
    const float* vs[3] = {xi0 + it, xi1 + it, xi2 + it};
    float acc = 0.0f;
#pragma unroll
    for (int s = 0; s < 3; ++s) {
        float4 a = ((const float4*)us[s])[l];
        float4 b = ((const float4*)vs[s])[l];
        acc += a.x * b.x + a.y * b.y + a.z * b.z + a.w * b.w;
    }
#pragma unroll
    for (int off = 8; off; off >>= 1) acc += __shfl_xor(acc, off, 32);
    if (l == 0) out[pair] = acc;
}

// ---------------------------------------------------------------- launcher

extern "C" void kernel_launch(void* const* d_in, const int* in_sizes, int n_in,
                              void* d_out, int out_size, void* d_ws, size_t ws_size,
                              hipStream_t stream) {
    const float* emb_u = (const float*)d_in[0];
    const float* emb_i = (const float*)d_in[1];
    const int* nid_u = (const int*)d_in[2];
    const int* nid_i = (const int*)d_in[3];
    const int* ei_u = (const int*)d_in[4];
    const int* ei_i = (const int*)d_in[5];
    const int* eli_u = (const int*)d_in[6];
    const int* eli_i = (const int*)d_in[7];
    const float* Wl_ui[2] = {(const float*)d_in[8],  (const float*)d_in[12]};
    const float* Wi_ui[2] = {(const float*)d_in[9],  (const float*)d_in[13]};
    const float* Wl_iu[2] = {(const float*)d_in[10], (const float*)d_in[14]};
    const float* Wi_iu[2] = {(const float*)d_in[11], (const float*)d_in[15]};

    const int NU = in_sizes[2];
    const int NI = in_sizes[3];
    const int E  = in_sizes[4];
    const int B  = in_sizes[6];

    float* ws = (float*)d_ws;
    size_t off = 0;
    float* xu[3];
    float* xi[3];
    for (int s = 0; s < 3; ++s) { xu[s] = ws + off; off += (size_t)NU * DD; }
    for (int s = 0; s < 3; ++s) { xi[s] = ws + off; off += (size_t)NI * DD; }
    float* ysrc = ws + off; off += (size_t)((NU > NI) ? NU : NI) * DD;
    float* degu = ws + off; off += NU;
    float* degi = ws + off; off += NI;  // contiguous with degu -> single zero pass
    float* wE = ws + off; off += E;
    float* out = (float*)d_out;

    const int T = 256;
    // degrees + edge weights
    k_zero<<<(NU + NI + T - 1) / T, T, 0, stream>>>(degu, NU + NI);
    k_degree<<<(E + T - 1) / T, T, 0, stream>>>(ei_u, ei_i, degu, degi, E);
    k_edge_w<<<(E + T - 1) / T, T, 0, stream>>>(ei_u, ei_i, degu, degi, wE, E);

    // stage-0 features (honor n_id gathers)
    k_gather_rows<<<((size_t)NU * DD + T - 1) / T, T, 0, stream>>>(emb_u, nid_u, xu[0], NU * DD);
    k_gather_rows<<<((size_t)NI * DD + T - 1) / T, T, 0, stream>>>(emb_i, nid_i, xi[0], NI * DD);

    dim3 gU((NU + 15) / 16, 4), gI((NI + 15) / 16, 4);
    int eBlocks = (E + ET - 1) / ET;

    for (int l = 0; l < 2; ++l) {
        // ---- cell user -> item (writes xi[l+1]) ----
        k_gemm64<<<gU, 32, 0, stream>>>(xu[l], Wl_ui[l], ysrc, NU);       // z_src rows
        k_gemm64<<<gI, 32, 0, stream>>>(xi[l], Wl_ui[l], xi[l + 1], NI);  // self term
        k_edge_msg<<<eBlocks, 128, 0, stream>>>(xu[l], xi[l], ei_u, ei_i,
                                                wE, ysrc, Wi_ui[l], xi[l + 1], E);
        k_lrelu<<<((size_t)NI * DD + T - 1) / T, T, 0, stream>>>(xi[l + 1], NI * DD);

        // ---- cell item -> user (writes xu[l+1]) ----
        k_gemm64<<<gI, 32, 0, stream>>>(xi[l], Wl_iu[l], ysrc, NI);
        k_gemm64<<<gU, 32, 0, stream>>>(xu[l], Wl_iu[l], xu[l + 1], NU);
        k_edge_msg<<<eBlocks, 128, 0, stream>>>(xi[l], xu[l], ei_i, ei_u,
                                                wE, ysrc, Wi_iu[l], xu[l + 1], E);
        k_lrelu<<<((size_t)NU * DD + T - 1) / T, T, 0, stream>>>(xu[l + 1], NU * DD);
    }

    // final pairwise dot over concatenated stages
    int pairsPerBlock = 256 / 16;
    k_dot<<<(B + pairsPerBlock - 1) / pairsPerBlock, 256, 0, stream>>>(
        xu[0], xu[1], xu[2], xi[0], xi[1], xi[2], eli_u, eli_i, out, B);
}